// Decoder_85057532330658
// MI455X (gfx1250) — compile-verified
//
#include <hip/hip_runtime.h>
#include <stdint.h>
#include <stddef.h>

// ---------------------------------------------------------------------------
// CDNA5 (gfx1250) implementation. fp32 WMMA 16x16x4 for all GEMM cores.
// Workload is gather/scatter bandwidth bound (AI ~ 4 FLOP/B << machine
// balance at 23.3 TB/s), so fp32 matrix math is the right precision: exact
// vs the reference, and compute is nowhere near the bottleneck.
// ---------------------------------------------------------------------------

typedef __attribute__((ext_vector_type(2))) float v2f;
typedef __attribute__((ext_vector_type(8))) float v8f;

constexpr int KC3 = 27;  // 3x3x3 conv offsets
constexpr int KC2 = 8;   // upsample factor

__device__ __forceinline__ float relu_f(float x) { return x > 0.f ? x : 0.f; }

// ----------------------------- elementwise ---------------------------------
__global__ void fill_bias_kernel(float* __restrict__ out, const float* __restrict__ b,
                                 int n, int d) {
  long i = (long)blockIdx.x * blockDim.x + threadIdx.x;
  if (i < (long)n * d) out[i] = b[i % d];
}

__global__ void relu_kernel(float* __restrict__ x, long tot) {
  long i = (long)blockIdx.x * blockDim.x + threadIdx.x;
  if (i < tot) x[i] = relu_f(x[i]);
}

__global__ void combine_residual_kernel(float* __restrict__ f, const float* __restrict__ h0,
                                        const float* __restrict__ h1, int n, int cout, int c2) {
  long i = (long)blockIdx.x * blockDim.x + threadIdx.x;
  if (i >= (long)n * cout) return;
  int row = (int)(i / cout), col = (int)(i % cout);
  float add = (col < c2) ? h0[(long)row * c2 + col] : h1[(long)row * c2 + (col - c2)];
  f[i] += add;
}

__global__ void rowscale_kernel(float* __restrict__ f, const float* __restrict__ mask,
                                int n, int c) {
  long i = (long)blockIdx.x * blockDim.x + threadIdx.x;
  if (i < (long)n * c) f[i] *= mask[i / c];
}

__global__ void mul_store_kernel(const float* __restrict__ a, const float* __restrict__ m,
                                 float* __restrict__ o, int n) {
  long i = (long)blockIdx.x * blockDim.x + threadIdx.x;
  if (i < n) o[i] = a[i] * m[i];
}

__global__ void bool2f_kernel(const unsigned char* __restrict__ t, float* __restrict__ o, int n) {
  long i = (long)blockIdx.x * blockDim.x + threadIdx.x;
  if (i < n) o[i] = t[i] ? 1.f : 0.f;
}

// --------------------------- gen_up (dense WMMA) ----------------------------
// g[(p*K2 + k)*cout + dd] = relu( sum_c f[p][c] * Wt[k][c][dd] + bt[dd] )
// One wave per 16-row x 16-col output tile; the 16-col tile always lies inside
// a single k-block since cout in {64,32,16}.
__global__ void __launch_bounds__(256)
genup_wmma_kernel(const float* __restrict__ f, const float* __restrict__ Wt,
                  const float* __restrict__ bt, float* __restrict__ g,
                  int nprev, int cin, int cout) {
  int wid  = blockIdx.x * 8 + (threadIdx.x >> 5);
  int lane = threadIdx.x & 31;
  int rowTiles = nprev >> 4;
  int colTiles = (KC2 * cout) >> 4;
  if (wid >= rowTiles * colTiles) return;
  int rt = wid / colTiles;
  int ct = wid - rt * colTiles;
  int j0  = ct * 16;
  int k   = j0 / cout;
  int dd0 = j0 - k * cout;
  int half = lane >> 4;     // lanes 16-31 carry K+2 for A/B, M+8 for C
  int l16  = lane & 15;
  int mrow = rt * 16 + l16; // A-matrix row M
  int col  = dd0 + l16;     // B/C column N
  v8f acc = {};
  for (int kk0 = 0; kk0 < cin; kk0 += 4) {
    int ka = kk0 + 2 * half;
    v2f a, b;
    a.x = f[(long)mrow * cin + ka];
    a.y = f[(long)mrow * cin + ka + 1];
    const float* wp = Wt + ((long)k * cin + ka) * cout + col;
    b.x = wp[0];
    b.y = wp[cout];
    acc = __builtin_amdgcn_wmma_f32_16x16x4_f32(false, a, false, b, (short)0, acc, false, false);
  }
  float bias = bt[col];
#pragma unroll
  for (int v = 0; v < 8; ++v) {
    int M = v + half * 8;
    long p = (long)rt * 16 + M;
    g[(p * KC2 + k) * cout + col] = relu_f(acc[v] + bias);
  }
}

// ---------------------- sparse conv3: gather-WMMA-scatter -------------------
// For each of 27 offsets k: y = f[km_in[k]] @ W[k]; out[km_out[k]] += y.
// out must be pre-initialized with the bias. One wave per (k, row-tile,
// col-tile). Columns >= d are zero-padded in B and skipped on store.
__global__ void __launch_bounds__(256)
conv3_scatter_wmma_kernel(const float* __restrict__ f, const float* __restrict__ W,
                          const int* __restrict__ km_in, const int* __restrict__ km_out,
                          float* __restrict__ out, int n, int c, int d) {
  int wid  = blockIdx.x * 8 + (threadIdx.x >> 5);
  int lane = threadIdx.x & 31;
  int rowTiles = n >> 4;
  int colTiles = (d + 15) >> 4;
  int tilesPerK = rowTiles * colTiles;
  if (wid >= KC3 * tilesPerK) return;
  int k   = wid / tilesPerK;
  int rem = wid - k * tilesPerK;
  int rt  = rem / colTiles;
  int ct  = rem - rt * colTiles;
  int half = lane >> 4;
  int l16  = lane & 15;
  int col  = ct * 16 + l16;
  int gRow = km_in[(long)k * n + rt * 16 + l16];  // gathered source row for M=l16
  v8f acc = {};
  for (int kk0 = 0; kk0 < c; kk0 += 4) {
    int ka = kk0 + 2 * half;
    v2f a = {}, b = {};
    if (ka < c)     a.x = f[(long)gRow * c + ka];
    if (ka + 1 < c) a.y = f[(long)gRow * c + ka + 1];
    if (col < d) {
      const float* wp = W + ((long)k * c + ka) * d + col;
      if (ka < c)     b.x = wp[0];
      if (ka + 1 < c) b.y = wp[d];
    }
    acc = __builtin_amdgcn_wmma_f32_16x16x4_f32(false, a, false, b, (short)0, acc, false, false);
  }
  if (col < d) {
#pragma unroll
    for (int v = 0; v < 8; ++v) {
      int M = v + half * 8;
      int oRow = km_out[(long)k * n + rt * 16 + M];
      atomicAdd(&out[(long)oRow * d + col], acc[v]);
    }
  }
}

// --------------------------- dense GEMM (WMMA) ------------------------------
// C(n,d) = A(n,c) @ B(c,d) + bias, optional relu. d may be < 16.
__global__ void __launch_bounds__(256)
gemm_bias_wmma_kernel(const float* __restrict__ A, const float* __restrict__ B,
                      const float* __restrict__ bias, float* __restrict__ C,
                      int n, int c, int d, int doRelu) {
  int wid  = blockIdx.x * 8 + (threadIdx.x >> 5);
  int lane = threadIdx.x & 31;
  int rowTiles = n >> 4;
  int colTiles = (d + 15) >> 4;
  if (wid >= rowTiles * colTiles) return;
  int rt = wid / colTiles;
  int ct = wid - rt * colTiles;
  int half = lane >> 4;
  int l16  = lane & 15;
  int m   = rt * 16 + l16;
  int col = ct * 16 + l16;
  v8f acc = {};
  for (int kk0 = 0; kk0 < c; kk0 += 4) {
    int ka = kk0 + 2 * half;
    v2f a = {}, b = {};
    if (ka < c)     a.x = A[(long)m * c + ka];
    if (ka + 1 < c) a.y = A[(long)m * c + ka + 1];
    if (col < d) {
      if (ka < c)     b.x = B[(long)ka * d + col];
      if (ka + 1 < c) b.y = B[(long)(ka + 1) * d + col];
    }
    acc = __builtin_amdgcn_wmma_f32_16x16x4_f32(false, a, false, b, (short)0, acc, false, false);
  }
  if (col < d) {
    float bv = bias[col];
#pragma unroll
    for (int v = 0; v < 8; ++v) {
      int M = v + half * 8;
      float val = acc[v] + bv;
      if (doRelu) val = relu_f(val);
      C[((long)rt * 16 + M) * d + col] = val;
    }
  }
}

// ------------------------- classifier conv (d == 1) -------------------------
__global__ void conv_cls_kernel(const float* __restrict__ f, const float* __restrict__ W,
                                const int* __restrict__ km_in, const int* __restrict__ km_out,
                                float* __restrict__ out, int n, int c) {
  long i = (long)blockIdx.x * blockDim.x + threadIdx.x;
  if (i >= (long)KC3 * n) return;
  int k = (int)(i / n);
  const float* w  = W + (long)k * c;
  const float* fr = f + (long)km_in[i] * c;
  float s = 0.f;
  for (int cc = 0; cc < c; ++cc) s += fr[cc] * w[cc];
  atomicAdd(&out[km_out[i]], s);
}

// ------------------------- max/argmax reduction -----------------------------
__global__ void reduce_max_pass1(const float* __restrict__ s, int n,
                                 float* __restrict__ bmax, int* __restrict__ barg) {
  __shared__ float smax[256];
  __shared__ int   sarg[256];
  int tid = threadIdx.x;
  long i = (long)blockIdx.x * blockDim.x + tid;
  float v = -3.402823466e38f;
  int   a = 0x7fffffff;
  if (i < n) { v = s[i]; a = (int)i; }
  smax[tid] = v; sarg[tid] = a;
  __syncthreads();
  for (int off = 128; off > 0; off >>= 1) {
    if (tid < off) {
      float v2 = smax[tid + off]; int a2 = sarg[tid + off];
      if (v2 > smax[tid] || (v2 == smax[tid] && a2 < sarg[tid])) { smax[tid] = v2; sarg[tid] = a2; }
    }
    __syncthreads();
  }
  if (tid == 0) { bmax[blockIdx.x] = smax[0]; barg[blockIdx.x] = sarg[0]; }
}

__global__ void reduce_max_pass2(const float* __restrict__ bmax, const int* __restrict__ barg,
                                 int nblocks, float* __restrict__ gmax, int* __restrict__ garg) {
  __shared__ float smax[256];
  __shared__ int   sarg[256];
  int tid = threadIdx.x;
  float v = -3.402823466e38f;
  int   a = 0x7fffffff;
  for (int i = tid; i < nblocks; i += 256) {
    float v2 = bmax[i]; int a2 = barg[i];
    if (v2 > v || (v2 == v && a2 < a)) { v = v2; a = a2; }
  }
  smax[tid] = v; sarg[tid] = a;
  __syncthreads();
  for (int off = 128; off > 0; off >>= 1) {
    if (tid < off) {
      float v2 = smax[tid + off]; int a2 = sarg[tid + off];
      if (v2 > smax[tid] || (v2 == smax[tid] && a2 < sarg[tid])) { smax[tid] = v2; sarg[tid] = a2; }
    }
    __syncthreads();
  }
  if (tid == 0) { gmax[0] = smax[0]; garg[0] = sarg[0]; }
}

__global__ void mask_kernel(const float* __restrict__ s, const float* __restrict__ gmax,
                            const int* __restrict__ garg, float* __restrict__ maskB,
                            float* __restrict__ kOut, int n) {
  long i = (long)blockIdx.x * blockDim.x + threadIdx.x;
  if (i >= n) return;
  float m = (gmax[0] < 0.f) ? ((i == garg[0]) ? 1.f : 0.f) : ((s[i] > 0.f) ? 1.f : 0.f);
  maskB[i] = m;
  kOut[i]  = m;
}

// ---------------------------------------------------------------------------
extern "C" void kernel_launch(void* const* d_in, const int* in_sizes, int n_in,
                              void* d_out, int out_size, void* d_ws, size_t ws_size,
                              hipStream_t stream) {
  (void)in_sizes; (void)n_in; (void)out_size; (void)ws_size;

  const float* x = (const float*)d_in[0];

  // params flattened in setup_inputs() dict insertion order per stage.
  struct SP {
    const float *Wt, *bt, *Wc, *bc, *W0a, *b0a, *W0b, *b0b,
                *W1a, *b1a, *W1b, *b1b, *W1c, *b1c, *Wcls, *bcls;
  } P[3];
  const int* KIn[3]; const int* KOut[3]; const unsigned char* Tb[3];
  {
    int idx = 1;
    for (int s = 0; s < 3; ++s) {
      const float** fp = (const float**)&P[s];
      for (int j = 0; j < 16; ++j) fp[j] = (const float*)d_in[idx++];
      KIn[s]  = (const int*)d_in[idx++];
      KOut[s] = (const int*)d_in[idx++];
      Tb[s]   = (const unsigned char*)d_in[idx++];
    }
  }

  const int  CH_[3]  = {64, 32, 16};
  const long NS_[3]  = {8192, 65536, 524288};
  const int  CIN_[3] = {128, 64, 32};
  const long NP_[3]  = {1024, 8192, 65536};

  // d_out layout: out, c0..c2, t0..t2, k0..k2 (flat, return order)
  float* o_base = (float*)d_out;
  float* o_out  = o_base;
  long off = NS_[2];
  float* o_c[3]; float* o_t[3]; float* o_k[3];
  for (int s = 0; s < 3; ++s) { o_c[s] = o_base + off; off += NS_[s]; }
  for (int s = 0; s < 3; ++s) { o_t[s] = o_base + off; off += NS_[s]; }
  for (int s = 0; s < 3; ++s) { o_k[s] = o_base + off; off += NS_[s]; }

  // workspace carve-up
  char* wsp = (char*)d_ws;
  size_t wo = 0;
  auto walloc = [&](size_t bytes) -> void* {
    void* p = wsp + wo; wo += (bytes + 255) & ~(size_t)255; return p;
  };
  float* bufU   = (float*)walloc(8388608ul * 4);  // n*cout (upsampled)
  float* bufF   = (float*)walloc(8388608ul * 4);  // n*cout (features)
  float* bufH0  = (float*)walloc(2097152ul * 4);  // n*c4
  float* bufH0b = (float*)walloc(4194304ul * 4);  // n*c2
  float* bufH1a = (float*)walloc(2097152ul * 4);  // n*c4
  float* bufH1b = (float*)walloc(2097152ul * 4);  // n*c4
  float* bufH1c = (float*)walloc(4194304ul * 4);  // n*c2
  float* maskB  = (float*)walloc(524288ul * 4);
  float* bmax   = (float*)walloc(2048ul * 4);
  int*   barg   = (int*)walloc(2048ul * 4);
  float* gmax   = (float*)walloc(256);
  int*   garg   = (int*)walloc(256);

  auto grid1 = [](long tot) { return dim3((unsigned)((tot + 255) / 256)); };
  auto gridw = [](long waves) { return dim3((unsigned)((waves + 7) / 8)); };

  auto conv3 = [&](const float* f, const float* W, const float* b,
                   const int* ki, const int* ko, float* out, long n, int c, int d) {
    fill_bias_kernel<<<grid1(n * d), 256, 0, stream>>>(out, b, (int)n, d);
    long waves = (long)KC3 * (n / 16) * ((d + 15) / 16);
    conv3_scatter_wmma_kernel<<<gridw(waves), 256, 0, stream>>>(f, W, ki, ko, out, (int)n, c, d);
  };

  const float* fin = x;
  for (int s = 0; s < 3; ++s) {
    long nprev = NP_[s]; int cin = CIN_[s]; int cout = CH_[s];
    long n = NS_[s];
    int c4 = cout / 4, c2 = cout / 2;
    const int* ki = KIn[s]; const int* ko = KOut[s];

    bool2f_kernel<<<grid1(n), 256, 0, stream>>>(Tb[s], o_t[s], (int)n);

    // gen_up + relu (fused into the WMMA epilogue)
    long guW = (nprev / 16) * (((long)KC2 * cout) / 16);
    genup_wmma_kernel<<<gridw(guW), 256, 0, stream>>>(fin, P[s].Wt, P[s].bt, bufU,
                                                      (int)nprev, cin, cout);
    // main conv + relu
    conv3(bufU, P[s].Wc, P[s].bc, ki, ko, bufF, n, cout, cout);
    relu_kernel<<<grid1(n * cout), 256, 0, stream>>>(bufF, n * cout);

    for (int r = 0; r < 3; ++r) {
      // branch 0: conv -> relu -> conv
      conv3(bufF, P[s].W0a + (long)r * KC3 * cout * c4, P[s].b0a + (long)r * c4,
            ki, ko, bufH0, n, cout, c4);
      relu_kernel<<<grid1(n * c4), 256, 0, stream>>>(bufH0, n * c4);
      conv3(bufH0, P[s].W0b + (long)r * KC3 * c4 * c2, P[s].b0b + (long)r * c2,
            ki, ko, bufH0b, n, c4, c2);
      // branch 1: gemm(relu) -> conv(relu) -> gemm
      long gw1 = (n / 16) * ((c4 + 15) / 16);
      gemm_bias_wmma_kernel<<<gridw(gw1), 256, 0, stream>>>(
          bufF, P[s].W1a + (long)r * cout * c4, P[s].b1a + (long)r * c4,
          bufH1a, (int)n, cout, c4, 1);
      conv3(bufH1a, P[s].W1b + (long)r * KC3 * c4 * c4, P[s].b1b + (long)r * c4,
            ki, ko, bufH1b, n, c4, c4);
      relu_kernel<<<grid1(n * c4), 256, 0, stream>>>(bufH1b, n * c4);
      long gw2 = (n / 16) * ((c2 + 15) / 16);
      gemm_bias_wmma_kernel<<<gridw(gw2), 256, 0, stream>>>(
          bufH1b, P[s].W1c + (long)r * c4 * c2, P[s].b1c + (long)r * c2,
          bufH1c, (int)n, c4, c2, 0);
      // concat + residual, in place on bufF
      combine_residual_kernel<<<grid1(n * cout), 256, 0, stream>>>(
          bufF, bufH0b, bufH1c, (int)n, cout, c2);
    }

    // classifier (d == 1) straight into d_out c-region
    fill_bias_kernel<<<grid1(n), 256, 0, stream>>>(o_c[s], P[s].bcls, (int)n, 1);
    conv_cls_kernel<<<grid1((long)KC3 * n), 256, 0, stream>>>(
        bufF, P[s].Wcls, ki, ko, o_c[s], (int)n, cout);

    // keep_mask
    int nb = (int)((n + 255) / 256);
    reduce_max_pass1<<<nb, 256, 0, stream>>>(o_c[s], (int)n, bmax, barg);
    reduce_max_pass2<<<1, 256, 0, stream>>>(bmax, barg, nb, gmax, garg);
    mask_kernel<<<grid1(n), 256, 0, stream>>>(o_c[s], gmax, garg, maskB, o_k[s], (int)n);

    // mask features for the next stage (in place)
    rowscale_kernel<<<grid1(n * cout), 256, 0, stream>>>(bufF, maskB, (int)n, cout);
    fin = bufF;
  }

  // final output: c2 * k2
  mul_store_kernel<<<grid1(NS_[2]), 256, 0, stream>>>(o_c[2], maskB, o_out, (int)NS_[2]);
}